// DetectionLoss_12652973654174
// MI455X (gfx1250) — compile-verified
//
#include <hip/hip_runtime.h>
#include <math.h>

// ---------------------------------------------------------------------------
// Detection loss for MI455X (gfx1250, wave32).
// Bandwidth/latency-bound (~9MB traffic, ~0.5 GFLOP): one thread per (b,a),
// per-image object tables staged in LDS, IoU recomputed (cheap) instead of
// cached, predictions loaded lazily for the sparse positives.
// Reductions: wave32 shfl butterflies + exact-f32 V_WMMA_F32_16X16X4_F32
// (A = 16 quantities x 4 packed partials, B = ones -> row totals), two-stage
// deterministic (block partials in workspace, single-wave finalize kernel).
// ---------------------------------------------------------------------------

#define TPB  256
#define NQ   8      // s_box, n_box, s_kp, n_kp, s_d, n_d, s_cls, n_posneg
#define MAXO 64
#define MAXK 10

typedef __attribute__((ext_vector_type(2))) float v2f;
typedef __attribute__((ext_vector_type(8))) float v8f;

#if defined(__has_builtin)
#if __has_builtin(__builtin_amdgcn_wmma_f32_16x16x4_f32)
#define HAVE_WMMA4 1
#endif
#endif
#ifndef HAVE_WMMA4
#define HAVE_WMMA4 0
#endif

__device__ __forceinline__ float smooth_l1(float d) {
  float ad = fabsf(d);
  return ad < 1.0f ? 0.5f * d * d : ad - 0.5f;
}

// torchvision sigmoid_focal_loss with inputs=t (one-hot), targets=p (logits),
// ALPHA=0.8, GAMMA=0.5 (pow(x,0.5) -> sqrtf, x in (0,1) since t in {0,1}).
__device__ __forceinline__ float focal_elem(float t, float p) {
  float prob    = 1.0f / (1.0f + expf(-t));
  float ce      = fmaxf(t, 0.0f) - t * p + log1pf(expf(-fabsf(t)));
  float p_t     = prob * p + (1.0f - prob) * (1.0f - p);
  float alpha_t = 0.8f * p + 0.2f * (1.0f - p);
  return alpha_t * ce * sqrtf(fmaxf(1.0f - p_t, 0.0f));
}

__device__ __forceinline__ float anchor_iou(float4 av, float area_a,
                                            float bx0, float by0,
                                            float bx1, float by1,
                                            float area_b, bool pad) {
  float ltx = fmaxf(av.x, bx0), lty = fmaxf(av.y, by0);
  float rbx = fminf(av.z, bx1), rby = fminf(av.w, by1);
  float wx = fmaxf(rbx - ltx, 0.0f), wy = fmaxf(rby - lty, 0.0f);
  float inter = wx * wy;
  float ov = inter / (area_a + area_b - inter);
  return pad ? -1.0f : ov;
}

__global__ __launch_bounds__(TPB) void det_loss_main(
    const float* __restrict__ pred_boxes,   // [B,A,4]
    const float* __restrict__ pred_classes, // [B,A,2]
    const float* __restrict__ pred_kp,      // [B,A,K2]
    const float* __restrict__ pred_depths,  // [B,A,1]
    const float* __restrict__ boxes,        // [B,O,4]
    const float* __restrict__ keypoints,    // [B,O,K2]
    const float* __restrict__ depths,       // [B,O,1]
    const float* __restrict__ anchors,      // [A,4]
    const int*   __restrict__ labels,       // [B,O,1]
    float* __restrict__ partials,           // [nblk,NQ]
    int B, int O, int A, int K2, int blocks_per_img)
{
  __shared__ float s_bx0[MAXO], s_by0[MAXO], s_bx1[MAXO], s_by1[MAXO];
  __shared__ float s_bcx[MAXO], s_bcy[MAXO], s_lbw[MAXO], s_lbh[MAXO];
  __shared__ float s_ab[MAXO], s_dep[MAXO];
  __shared__ float s_kpv[MAXO][MAXK];
  __shared__ int   s_kpm[MAXO], s_dok[MAXO], s_lab[MAXO];
  __shared__ float wsum[TPB / 32][NQ];

  const int tid = threadIdx.x;
  const int b   = blockIdx.x / blocks_per_img;   // A % TPB == 0 -> uniform b
  const int a   = (blockIdx.x % blocks_per_img) * TPB + tid;
  const int K   = K2 > MAXK ? MAXK : K2;

  // ---- stage this image's object tables into LDS --------------------------
  for (int o = tid; o < O && o < MAXO; o += TPB) {
    const float* bp = boxes + (size_t)(b * O + o) * 4;
    float x0 = bp[0], y0 = bp[1], x1 = bp[2], y1 = bp[3];
    s_bx0[o] = x0; s_by0[o] = y0; s_bx1[o] = x1; s_by1[o] = y1;
    float bw = x1 - x0, bh = y1 - y0;
    s_bcx[o] = (x0 + x1) * 0.5f;
    s_bcy[o] = (y0 + y1) * 0.5f;
    s_lbw[o] = logf(bw > 0.0f ? bw : 1.0f);   // log(bwh_safe)
    s_lbh[o] = logf(bh > 0.0f ? bh : 1.0f);
    s_ab[o]  = bw * bh;
    s_lab[o] = labels[b * O + o];
    int m = 0;
    for (int j = 0; j < K; ++j) {
      float v  = keypoints[(size_t)(b * O + o) * K2 + j];
      bool nan = (v != v);
      if (!nan) m |= (1 << j);
      s_kpv[o][j] = nan ? 0.0f : v;
    }
    s_kpm[o] = m;
    float dv = depths[b * O + o];
    s_dok[o] = (dv == dv) ? 1 : 0;
    s_dep[o] = (dv == dv) ? dv : 0.0f;
  }
  __syncthreads();

  // ---- per-anchor precompute ---------------------------------------------
  const float4 av = ((const float4*)anchors)[a];
  const float acx = (av.x + av.z) * 0.5f, acy = (av.y + av.w) * 0.5f;
  const float aw  = av.z - av.x,          ah  = av.w - av.y;
  const float area_a = aw * ah;
  const float inv_vw = 1.0f / (0.1f * aw);   // 1/(V0*awh)
  const float inv_vh = 1.0f / (0.1f * ah);
  const float law = logf(aw), lah = logf(ah);

  // nearly every anchor is a negative -> warm up classes early
  __builtin_prefetch(pred_classes + ((size_t)b * A + a) * 2, 0, 1);

  // ---- pass 1: max IoU over objects --------------------------------------
  float maxov = -INFINITY;
  for (int o = 0; o < O; ++o) {
    float ov = anchor_iou(av, area_a, s_bx0[o], s_by0[o], s_bx1[o], s_by1[o],
                          s_ab[o], s_lab[o] < 0);
    maxov = fmaxf(maxov, ov);
  }
  const bool neg = maxov < 0.1f;     // NEG_TH

  float acc[NQ];
#pragma unroll
  for (int q = 0; q < NQ; ++q) acc[q] = 0.0f;

  const float2 pc = ((const float2*)pred_classes)[(size_t)b * A + a];

  // ---- pass 2: positive contributions (recompute IoU, lazy pred loads) ---
  bool  loaded = false;
  float4 pb = make_float4(0.f, 0.f, 0.f, 0.f);
  float pkp[MAXK];
  float pd = 0.0f;

  for (int o = 0; o < O; ++o) {
    float ov = anchor_iou(av, area_a, s_bx0[o], s_by0[o], s_bx1[o], s_by1[o],
                          s_ab[o], s_lab[o] < 0);
    bool pos = (fabsf(maxov - ov) < 1e-6f) && (ov > 0.5f);   // POS_TH
    if (!pos) continue;

    if (!loaded) {
      loaded = true;
      pb = ((const float4*)pred_boxes)[(size_t)b * A + a];
      const float* kpp = pred_kp + ((size_t)b * A + a) * K2;
      for (int j = 0; j < K; ++j) pkp[j] = kpp[j];
      pd = pred_depths[(size_t)b * A + a];
    }

    // boxes: SSD encode, smooth-L1 over 4 components
    float enc[4];
    enc[0] = (s_bcx[o] - acx) * inv_vw;
    enc[1] = (s_bcy[o] - acy) * inv_vh;
    enc[2] = (s_lbw[o] - law) * 5.0f;   // /V1
    enc[3] = (s_lbh[o] - lah) * 5.0f;
    float pv[4] = {pb.x, pb.y, pb.z, pb.w};
#pragma unroll
    for (int c = 0; c < 4; ++c) {
      if (!(enc[c] != enc[c])) {        // ~isnan(enc)
        acc[0] += smooth_l1(pv[c] - enc[c]);
        acc[1] += 1.0f;
      }
    }

    // keypoints: per-component NaN mask
    int m = s_kpm[o];
    for (int j = 0; j < K; ++j) {
      if ((m >> j) & 1) {
        float ek = ((j & 1) ? (s_kpv[o][j] - acy) * inv_vh
                            : (s_kpv[o][j] - acx) * inv_vw);
        acc[2] += smooth_l1(pkp[j] - ek);
        acc[3] += 1.0f;
      }
    }

    // depth: identity encoding
    if (s_dok[o]) {
      acc[4] += smooth_l1(pd - s_dep[o]);
      acc[5] += 1.0f;
    }

    // classes: focal over one-hot(clip(label,0,1))
    int lc = s_lab[o]; lc = lc < 0 ? 0 : (lc > 1 ? 1 : lc);
    float t0 = (lc == 0) ? 1.0f : 0.0f;
    float t1 = (lc == 1) ? 1.0f : 0.0f;
    acc[6] += focal_elem(t0, pc.x) + focal_elem(t1, pc.y);
    acc[7] += 1.0f;
  }

  if (neg) {   // negatives vs label (1,0)
    acc[6] += focal_elem(1.0f, pc.x) + focal_elem(0.0f, pc.y);
    acc[7] += 1.0f;
  }

  // ---- block reduction: wave32 butterfly, then one f32 WMMA ---------------
#pragma unroll
  for (int off = 16; off > 0; off >>= 1) {
#pragma unroll
    for (int q = 0; q < NQ; ++q)
      acc[q] += __shfl_xor(acc[q], off, 32);
  }
  const int lane = tid & 31, wv = tid >> 5;
  if (lane == 0) {
#pragma unroll
    for (int q = 0; q < NQ; ++q) wsum[wv][q] = acc[q];
  }
  __syncthreads();

  if (tid < 32) {
#if HAVE_WMMA4
    // A (16x4 f32): row q = quantity, A[q][k] = wsum[2k][q] + wsum[2k+1][q].
    // Lane l<16 supplies K=0,1 (VGPR0,1); lane>=16 supplies K=2,3.
    int q = tid & 15, half = tid >> 4;
    float a0 = 0.0f, a1 = 0.0f;
    if (q < NQ) {
      int k0 = 2 * half;
      a0 = wsum[2 * k0][q]     + wsum[2 * k0 + 1][q];
      a1 = wsum[2 * k0 + 2][q] + wsum[2 * k0 + 3][q];
    }
    v2f Am; Am[0] = a0;   Am[1] = a1;
    v2f Bm; Bm[0] = 1.0f; Bm[1] = 1.0f;   // ones -> D[q][n] = sum_k A[q][k]
    v8f Cm = {};
    v8f Dm = __builtin_amdgcn_wmma_f32_16x16x4_f32(
        false, Am, false, Bm, (short)0, Cm, false, false);
    if (tid == 0) {   // lane 0: D VGPR j = row M=j, N=0 -> block total q=j
      float* pp = partials + (size_t)blockIdx.x * NQ;
#pragma unroll
      for (int j = 0; j < NQ; ++j) pp[j] = Dm[j];
    }
#else
    if (tid < NQ) {
      float s = 0.0f;
      for (int w = 0; w < TPB / 32; ++w) s += wsum[w][tid];
      partials[(size_t)blockIdx.x * NQ + tid] = s;
    }
#endif
  }
}

// One wave reduces all block partials (deterministic order) + finalizes.
__global__ __launch_bounds__(32) void det_loss_final(
    const float* __restrict__ partials, int nblk, float* __restrict__ out)
{
  __shared__ float tmp[32][NQ];
  const int l = threadIdx.x;

  float acc[NQ];
#pragma unroll
  for (int q = 0; q < NQ; ++q) acc[q] = 0.0f;
  for (int i = l; i < nblk; i += 32) {
#pragma unroll
    for (int q = 0; q < NQ; ++q) acc[q] += partials[(size_t)i * NQ + q];
  }
#pragma unroll
  for (int q = 0; q < NQ; ++q) tmp[l][q] = acc[q];
  __syncthreads();

  float tot[NQ];
#if HAVE_WMMA4
  // A[q][k] = sum_{j<8} tmp[8k+j][q]; B = ones -> D row q = grand total.
  int q = l & 15, half = l >> 4;
  float a0 = 0.0f, a1 = 0.0f;
  if (q < NQ) {
    int base = 16 * half;
#pragma unroll
    for (int j = 0; j < 8; ++j) { a0 += tmp[base + j][q]; a1 += tmp[base + 8 + j][q]; }
  }
  v2f Am; Am[0] = a0;   Am[1] = a1;
  v2f Bm; Bm[0] = 1.0f; Bm[1] = 1.0f;
  v8f Cm = {};
  v8f Dm = __builtin_amdgcn_wmma_f32_16x16x4_f32(
      false, Am, false, Bm, (short)0, Cm, false, false);
#pragma unroll
  for (int j = 0; j < NQ; ++j) tot[j] = Dm[j];   // valid on lanes 0..15
#else
#pragma unroll
  for (int j = 0; j < NQ; ++j) {
    float s = 0.0f;
    for (int i = 0; i < 32; ++i) s += tmp[i][j];
    tot[j] = s;
  }
#endif

  if (l == 0) {
    float s_box = tot[0], n_box = tot[1];
    float s_kp  = tot[2], n_kp  = tot[3];
    float s_d   = tot[4], n_d   = tot[5];
    float s_c   = tot[6], n_pn  = tot[7];
    float lb = n_box > 0.0f ? s_box / (n_box * n_box) : 0.0f;
    float lk = n_kp  > 0.0f ? s_kp  / (n_kp  * n_kp ) : 0.0f;
    float ld = n_d   > 0.0f ? s_d   / (n_d   * n_d  ) : 0.0f;
    float cc = 2.0f * n_pn;
    float lc = cc > 0.0f ? s_c / (cc * cc) : 0.0f;
    out[0] = lb; out[1] = lk; out[2] = ld; out[3] = lc;
    out[4] = lb + lk + ld + lc;
  }
}

extern "C" void kernel_launch(void* const* d_in, const int* in_sizes, int n_in,
                              void* d_out, int out_size, void* d_ws, size_t ws_size,
                              hipStream_t stream) {
  const float* pred_boxes   = (const float*)d_in[0];
  const float* pred_classes = (const float*)d_in[1];
  const float* pred_kp      = (const float*)d_in[2];
  const float* pred_depths  = (const float*)d_in[3];
  const float* boxes        = (const float*)d_in[4];
  const float* keypoints    = (const float*)d_in[5];
  const float* depths       = (const float*)d_in[6];
  const float* anchors      = (const float*)d_in[7];
  const int*   labels       = (const int*)d_in[8];

  const int A  = in_sizes[7] / 4;                 // anchors [A,4]
  const int B  = in_sizes[0] / (A * 4);           // pred_boxes [B,A,4]
  const int O  = in_sizes[8] / B;                 // labels [B,O,1]
  const int K2 = in_sizes[5] / (B * O);           // keypoints [B,O,K2]

  const int blocks_per_img = A / TPB;             // A=16384 -> 64
  const int nblk = B * blocks_per_img;            // 512

  float* partials = (float*)d_ws;                 // nblk*NQ floats

  det_loss_main<<<nblk, TPB, 0, stream>>>(
      pred_boxes, pred_classes, pred_kp, pred_depths,
      boxes, keypoints, depths, anchors, labels,
      partials, B, O, A, K2, blocks_per_img);

  det_loss_final<<<1, 32, 0, stream>>>(partials, nblk, (float*)d_out);
}